// TransNonlinear_11089605558939
// MI455X (gfx1250) — compile-verified
//
#include <hip/hip_runtime.h>

typedef __attribute__((ext_vector_type(2))) float v2f;
typedef __attribute__((ext_vector_type(4))) float v4f;
typedef __attribute__((ext_vector_type(8))) float v8f;

#define BATCH      16384
#define N_NEURONS  4096
#define BS         128     // neuron block size
#define NB         32      // number of diagonal blocks
#define ROW_TILES  8       // 8 row-tiles of 16 => 128 batch rows per workgroup
#define LDS_STRIDE 132     // 128 + 4 pad floats: conflict-free ds_load_b64 reads

__global__ __launch_bounds__(256)
void blockdiag_relu_wmma(const float* __restrict__ v,
                         const float* __restrict__ dx,
                         const float* __restrict__ A,    // [NB][BS][BS]
                         const float* __restrict__ Bm,   // [N_NEURONS][2]
                         const float* __restrict__ bptr, // scalar bias
                         float* __restrict__ out)
{
    __shared__ float vt[16][LDS_STRIDE];   // one 16x128 v tile
    __shared__ float dxs[128 * 2];         // dx for the 128 rows of this WG

    const int nb   = blockIdx.y;                 // which diagonal block
    const int row0 = blockIdx.x * (16 * ROW_TILES);
    const int tid  = threadIdx.x;
    const int lane = tid & 31;
    const int wave = tid >> 5;                   // 0..7 -> 16-col stripe

    const int m = lane & 15;                     // row index of A-operand / col index of B-operand
    const int h = lane >> 4;                     // K half: lanes 16-31 hold K+2/K+3

    // dx for all 128 rows handled by this workgroup -> LDS (256 floats)
    dxs[tid] = dx[(size_t)row0 * 2 + tid];

    const int ncol = wave * 16 + m;              // column within the 128-wide block
    const int gcol = nb * BS + ncol;             // global output column

    const float bias = bptr[0];
    const float bm0  = Bm[gcol * 2 + 0];
    const float bm1  = Bm[gcol * 2 + 1];

    // Cache this wave's A column-stripe in registers:
    // breg[s] = { A[nb][4s+2h][ncol], A[nb][4s+2h+1][ncol] }   (B-operand layout)
    const float* Abase = A + (size_t)nb * BS * BS + (size_t)(2 * h) * BS + ncol;
    v2f breg[32];
#pragma unroll
    for (int s = 0; s < 32; ++s) {
        breg[s].x = Abase[(size_t)(4 * s) * BS];
        breg[s].y = Abase[(size_t)(4 * s) * BS + BS];
    }

    for (int rt = 0; rt < ROW_TILES; ++rt) {
        const int rbase = row0 + rt * 16;

        __syncthreads();   // prior tile's readers done (also fences dxs on rt==0)

        // Cooperative coalesced load of v tile [16][128] via float4:
        // 512 quads, 256 threads -> 2 each; 32 consecutive lanes cover one row.
#pragma unroll
        for (int i = 0; i < 2; ++i) {
            const int idx = tid + i * 256;       // 0..511
            const int r   = idx >> 5;            // 0..15
            const int q   = idx & 31;            // 0..31 (float4 within row)
            v4f val = *(const v4f*)(v + (size_t)(rbase + r) * N_NEURONS + nb * BS + q * 4);
            *(v4f*)&vt[r][q * 4] = val;
        }
        __syncthreads();

        v8f c = {0.f, 0.f, 0.f, 0.f, 0.f, 0.f, 0.f, 0.f};
#pragma unroll
        for (int s = 0; s < 32; ++s) {
            // A-operand: a = { v[m, 4s+2h], v[m, 4s+2h+1] }  (8B-aligned LDS read)
            v2f a = *(const v2f*)&vt[m][4 * s + 2 * h];
            c = __builtin_amdgcn_wmma_f32_16x16x4_f32(
                    /*neg_a=*/false, a, /*neg_b=*/false, breg[s],
                    /*c_mod=*/(short)0, c, /*reuse_a=*/false, /*reuse_b=*/false);
        }

        // Epilogue: c[r] maps to out[rbase + r + 8h][gcol]
#pragma unroll
        for (int r = 0; r < 8; ++r) {
            const int lrow = rt * 16 + r + 8 * h;      // row within this WG's 128 rows
            const float d0 = dxs[lrow * 2 + 0];
            const float d1 = dxs[lrow * 2 + 1];
            float val = c[r] + d0 * bm0 + d1 * bm1 + bias;
            val = val > 0.f ? val : 0.f;
            out[(size_t)(row0 + lrow) * N_NEURONS + gcol] = val;
        }
    }
}

extern "C" void kernel_launch(void* const* d_in, const int* in_sizes, int n_in,
                              void* d_out, int out_size, void* d_ws, size_t ws_size,
                              hipStream_t stream) {
    (void)in_sizes; (void)n_in; (void)d_ws; (void)ws_size;
    const float* v    = (const float*)d_in[0];   // [16384, 4096]
    const float* dx   = (const float*)d_in[1];   // [16384, 2]
    const float* A    = (const float*)d_in[2];   // [32, 128, 128]
    const float* Bm   = (const float*)d_in[3];   // [4096, 2]
    const float* bias = (const float*)d_in[4];   // scalar
    float* out        = (float*)d_out;           // [16384, 4096]
    (void)out_size;

    dim3 grid(BATCH / (16 * ROW_TILES), NB);     // (128, 32)
    dim3 block(256);
    blockdiag_relu_wmma<<<grid, block, 0, stream>>>(v, dx, A, Bm, bias, out);
}